// Trace_Block_34153579938666
// MI455X (gfx1250) — compile-verified
//
#include <hip/hip_runtime.h>
#include <cstdint>
#include <cstddef>

// ---------------------------------------------------------------------------
// 8 batches x 20x20 pixels = 3200 seqs, L=120, in_c=3,
// bi hid=50 (gates 200), uni hid=10 (gates 40).
// ---------------------------------------------------------------------------
#define BSEQ 3200
#define LSEQ 120

typedef __attribute__((ext_vector_type(2))) float v2f;
typedef __attribute__((ext_vector_type(4))) float v4f;
typedef __attribute__((ext_vector_type(8))) float v8f;

#if __has_builtin(__builtin_amdgcn_global_load_async_to_lds_b128)
#define LSTM_ASYNC_XG 1
typedef __attribute__((ext_vector_type(4))) int v4i;
typedef __attribute__((address_space(1))) v4i g_v4i;   // global int4*
typedef __attribute__((address_space(3))) v4i l_v4i;   // LDS int4*
#endif

__device__ __forceinline__ float sigm_(float x) { return 1.f / (1.f + __expf(-x)); }

// ---------------------------------------------------------------------------
// x (8,120,3,20,20) -> xs4 (3200, 120, 4): seq-major, K padded 3->4 with zero.
// ---------------------------------------------------------------------------
__global__ __launch_bounds__(256) void reorder_x_kernel(const float* __restrict__ x,
                                                        float* __restrict__ xs4) {
  int idx = blockIdx.x * 256 + threadIdx.x;
  if (idx >= 8 * LSEQ * 400) return;
  int p = idx % 400;
  int t = (idx / 400) % LSEQ;
  int n = idx / (400 * LSEQ);
  const float* xb = x + ((size_t)(n * LSEQ + t) * 3) * 400 + p;
  v4f v;
  v.x = xb[0];
  v.y = xb[400];
  v.z = xb[800];
  v.w = 0.f;
  *(v4f*)(xs4 + (size_t)(n * 400 + p) * LSEQ * 4 + (size_t)t * 4) = v;
}

// ---------------------------------------------------------------------------
// Pack W (G,K) row-major into zero-padded, K-pair-interleaved transpose:
//   WTp[(k/2)*GP + n] = float2{ W[n][k], W[n][k+1] }
// Makes the WMMA B-operand a single contiguous 8B load.
// ---------------------------------------------------------------------------
__global__ __launch_bounds__(256) void pack_wt_kernel(const float* __restrict__ W,
                                                      float* __restrict__ WTp,
                                                      int K, int KP, int G, int GP) {
  int idx = blockIdx.x * 256 + threadIdx.x;
  if (idx >= (KP / 2) * GP) return;
  int k2 = idx / GP, n = idx % GP;
  int k = 2 * k2;
  WTp[idx * 2]     = (k < K && n < G) ? W[n * K + k] : 0.f;
  WTp[idx * 2 + 1] = (k + 1 < K && n < G) ? W[n * K + k + 1] : 0.f;
}

__global__ __launch_bounds__(256) void fill_zero_kernel(float* __restrict__ p, int nelem) {
  int idx = blockIdx.x * 256 + threadIdx.x;
  if (idx < nelem) p[idx] = 0.f;
}

// ---------------------------------------------------------------------------
// Y[M,G] = X[M,KS] @ W^T + (bih+bhh)   -- fp32 WMMA 16x16x4, K fully unrolled.
// Inner loop: two unguarded b64 loads + one v_wmma (all operands pre-padded).
// ---------------------------------------------------------------------------
template <int KS, int GP, int NT, int G>
__global__ __launch_bounds__(256) void gemm_wmma_kernel(
    const float* __restrict__ X, const float* __restrict__ WTp,
    const float* __restrict__ bih, const float* __restrict__ bhh,
    float* __restrict__ Y, int M) {
  const int tid = threadIdx.x;
  const int wid = tid >> 5, lane = tid & 31;
  const int half = lane >> 4, l15 = lane & 15;

  int tile = blockIdx.x * 8 + wid;        // 8 waves/block, one 16x16 tile each
  int mtiles = M >> 4;
  int mt = tile / NT, nt = tile % NT;     // wave-uniform -> EXEC all ones
  if (mt >= mtiles) return;

  const float* xr = X + (size_t)(mt * 16 + l15) * KS + half * 2;
  const int n = nt * 16 + l15;
  const float* wr = WTp + ((size_t)half * GP + n) * 2;

  v8f acc = {};
#pragma unroll
  for (int kb = 0; kb < KS; kb += 4) {
    v2f av = *(const v2f*)(xr + kb);
    v2f bv = *(const v2f*)(wr + (size_t)kb * GP);   // (kb/2)*GP*2
    acc = __builtin_amdgcn_wmma_f32_16x16x4_f32(false, av, false, bv,
                                                (short)0, acc, false, false);
  }
  if (n < G) {
    float bias = bih[n] + bhh[n];
    int mrow = mt * 16 + half * 8;
#pragma unroll
    for (int r = 0; r < 8; ++r)
      Y[(size_t)(mrow + r) * G + n] = acc[r] + bias;
  }
}

// ---------------------------------------------------------------------------
// Batched LSTM recurrence: one workgroup owns 16 sequences for all 120 steps.
//   H hidden (50/10)  G=4H (200/40)  KP=pad4(H) (52/12)  NT=ceil(G/16)  GP=NT*16
// h, c, gates, Whh^T (pair-packed) resident in LDS for the whole time loop.
// xg(t) is double-buffered in LDS via GLOBAL_LOAD_ASYNC_TO_LDS_B128: step t+1's
// rows stream in (ASYNCcnt) while step t's gate GEMM runs on WMMA.
// ---------------------------------------------------------------------------
template <int H, int G, int KP, int NT, int GP>
__global__ __launch_bounds__(256) void lstm_recur_kernel(
    const float* __restrict__ xg,   // (BSEQ, LSEQ, G)
    const float* __restrict__ Whh,  // (G, H) row-major
    float* __restrict__ hout,       // (BSEQ, LSEQ, OS)
    int OS, int coff, int rev) {
  __shared__ float sWT[KP * GP];    // pair-packed Whh^T, zero padded
  __shared__ float sH[16 * KP];     // h tile, zero-padded cols
  __shared__ float sC[16 * H];      // c tile
  __shared__ float sG[16 * GP];     // gate accumulators
#ifdef LSTM_ASYNC_XG
  constexpr int NCH = G / 4;                    // b128 chunks per row
  constexpr int NB4 = 16 * NCH;                 // valid chunks per step
  constexpr int NBP = (NB4 + 255) & ~255;       // padded -> uniform batches
  constexpr int NBI = NBP / 256;                // async instrs per wave per step
  __shared__ float sX[2][NBP * 4];              // double-buffered xg(t)
#endif

  const int tid = threadIdx.x;
  const int b0 = blockIdx.x * 16;

  for (int i = tid; i < (KP / 2) * GP; i += 256) {
    int k2 = i / GP, nn = i % GP;
    int k = 2 * k2;
    sWT[i * 2]     = (k < H && nn < G) ? Whh[nn * H + k] : 0.f;
    sWT[i * 2 + 1] = (k + 1 < H && nn < G) ? Whh[nn * H + k + 1] : 0.f;
  }
  for (int i = tid; i < 16 * KP; i += 256) sH[i] = 0.f;
  for (int i = tid; i < 16 * H; i += 256) sC[i] = 0.f;
  __syncthreads();

  const int wid = tid >> 5, lane = tid & 31;
  const int half = lane >> 4, l15 = lane & 15;

#ifdef LSTM_ASYNC_XG
  // Issue one uniform batch of async b128 loads for timestep `step` into buf.
  auto stage = [&](int buf, int step) {
#pragma unroll
    for (int kk = 0; kk < NBI; ++kk) {
      int k = tid + kk * 256;                 // padded chunks read harmless data
      int m = k / NCH, j4 = (k % NCH) * 4;
      const float* src = xg + ((size_t)(b0 + m) * LSEQ + step) * G + j4;
      __builtin_amdgcn_global_load_async_to_lds_b128(
          (g_v4i*)src, (l_v4i*)&sX[buf][k * 4], 0, 0);
    }
  };
  stage(0, rev ? (LSEQ - 1) : 0);             // preload step 0
#endif

  for (int t = 0; t < LSEQ; ++t) {
    const int tt = rev ? (LSEQ - 1 - t) : t;

#ifdef LSTM_ASYNC_XG
    // Kick off step t+1's xg transfer; it overlaps the whole WMMA phase.
    {
      int tn = (t + 1 < LSEQ) ? (t + 1) : t;  // last iter: re-stage (harmless)
      stage((t + 1) & 1, rev ? (LSEQ - 1 - tn) : tn);
    }
#else
    if (tid < 16) {
      int tn = rev ? (tt - 1) : (tt + 1);
      if (tn >= 0 && tn < LSEQ)
        __builtin_prefetch(&xg[((size_t)(b0 + tid) * LSEQ + tn) * G], 0, 3);
    }
#endif

    // gates[16,G] = h[16,H] @ Whh^T[H,G]  (wave-uniform; EXEC all ones)
    for (int nt = wid; nt < NT; nt += 8) {
      const int col = nt * 16 + l15;
      const float* hr = &sH[l15 * KP + half * 2];
      const float* wr = &sWT[((size_t)half * GP + col) * 2];
      v8f acc = {};
#pragma unroll
      for (int kb = 0; kb < KP; kb += 4) {
        v2f av = *(const v2f*)(hr + kb);
        v2f bv = *(const v2f*)(wr + (size_t)kb * GP);
        acc = __builtin_amdgcn_wmma_f32_16x16x4_f32(false, av, false, bv,
                                                    (short)0, acc, false, false);
      }
      const int mrow = half * 8;
#pragma unroll
      for (int r = 0; r < 8; ++r) sG[(mrow + r) * GP + col] = acc[r];
    }

#ifdef LSTM_ASYNC_XG
    // In-order completion: depth NBI == exactly the t+1 batch still in flight,
    // so step t's batch is guaranteed landed in sX[t&1].
#if __has_builtin(__builtin_amdgcn_s_wait_asynccnt)
    __builtin_amdgcn_s_wait_asynccnt(NBI);
#else
    asm volatile("s_wait_asynccnt %0" ::"n"(NBI));
#endif
#endif
    __syncthreads();

    // Activation / state update: gate order i,f,g,o (jnp.split order).
    for (int i = tid; i < 16 * H; i += 256) {
      int m = i / H, j = i % H;
#ifdef LSTM_ASYNC_XG
      const float* xr = &sX[t & 1][m * G];
#else
      const float* xr = xg + ((size_t)(b0 + m) * LSEQ + tt) * G;
#endif
      float gi = sG[m * GP + j] + xr[j];
      float gf = sG[m * GP + j + H] + xr[j + H];
      float gg = sG[m * GP + j + 2 * H] + xr[j + 2 * H];
      float go = sG[m * GP + j + 3 * H] + xr[j + 3 * H];
      float c = sigm_(gf) * sC[i] + sigm_(gi) * tanhf(gg);
      float h = sigm_(go) * tanhf(c);
      sC[i] = c;
      sH[m * KP + j] = h;
      hout[((size_t)(b0 + m) * LSEQ + tt) * OS + coff + j] = h;
    }
    __syncthreads();
  }
}

// ---------------------------------------------------------------------------
// hu2 (3200,120,10) last step -> out (8,10,20,20)
// ---------------------------------------------------------------------------
__global__ __launch_bounds__(256) void extract_out_kernel(const float* __restrict__ hu,
                                                          float* __restrict__ out) {
  int idx = blockIdx.x * 256 + threadIdx.x;
  if (idx >= 8 * 10 * 400) return;
  int p = idx % 400;
  int ch = (idx / 400) % 10;
  int n = idx / 4000;
  out[idx] = hu[((size_t)(n * 400 + p) * LSEQ + (LSEQ - 1)) * 10 + ch];
}

// ---------------------------------------------------------------------------
extern "C" void kernel_launch(void* const* d_in, const int* in_sizes, int n_in,
                              void* d_out, int out_size, void* d_ws, size_t ws_size,
                              hipStream_t stream) {
  (void)in_sizes; (void)n_in; (void)out_size; (void)ws_size;
  const float* x = (const float*)d_in[0];
  // params flatten: 6 layer dicts x {Wih, Whh, bih, bhh}
  auto P = [&](int g, int j) { return (const float*)d_in[1 + g * 4 + j]; };
  // g: 0=bi0_f 1=bi0_b 2=bi1_f 3=bi1_b 4=uni0 5=uni1

  float* ws = (float*)d_ws;
  float* xs4   = ws;                                    //  1,536,000 f
  float* xg    = xs4 + (size_t)BSEQ * LSEQ * 4;         // 76,800,000 f (reused)
  float* hcat1 = xg + (size_t)BSEQ * LSEQ * 200;        // 38,400,000 f
  float* hcat2 = hcat1 + (size_t)BSEQ * LSEQ * 100;     // 38,400,000 f
  float* hu1p  = hcat2 + (size_t)BSEQ * LSEQ * 100;     //  4,608,000 f (stride 12, zeroed)
  float* hu2   = hu1p + (size_t)BSEQ * LSEQ * 12;       //  3,840,000 f
  float* wtp   = hu2 + (size_t)BSEQ * LSEQ * 10;        //     20,800 f (reused)

  const int M = BSEQ * LSEQ;            // 384000 rows, divisible by 16
  const int mtiles = M / 16;

  auto pack = [&](int g, int K, int KP, int G, int GP) {
    int n = (KP / 2) * GP;
    pack_wt_kernel<<<(n + 255) / 256, 256, 0, stream>>>(P(g, 0), wtp, K, KP, G, GP);
  };

  // 0) layout reorder + zero the padded uni0 h-buffer
  reorder_x_kernel<<<(8 * LSEQ * 400 + 255) / 256, 256, 0, stream>>>(x, xs4);
  fill_zero_kernel<<<(BSEQ * LSEQ * 12 + 255) / 256, 256, 0, stream>>>(hu1p, BSEQ * LSEQ * 12);

  const int blk_bi = (mtiles * 13 + 7) / 8;
  const int blk_un = (mtiles * 3 + 7) / 8;

  // 1) bi0 forward / backward  (in 3 pad 4 -> hid 50), halves of hcat1
  pack(0, 3, 4, 200, 208);
  gemm_wmma_kernel<4, 208, 13, 200><<<blk_bi, 256, 0, stream>>>(xs4, wtp, P(0, 2), P(0, 3), xg, M);
  lstm_recur_kernel<50, 200, 52, 13, 208><<<BSEQ / 16, 256, 0, stream>>>(xg, P(0, 1), hcat1, 100, 0, 0);
  pack(1, 3, 4, 200, 208);
  gemm_wmma_kernel<4, 208, 13, 200><<<blk_bi, 256, 0, stream>>>(xs4, wtp, P(1, 2), P(1, 3), xg, M);
  lstm_recur_kernel<50, 200, 52, 13, 208><<<BSEQ / 16, 256, 0, stream>>>(xg, P(1, 1), hcat1, 100, 50, 1);

  // 2) bi1 forward / backward  (in 100 -> hid 50), halves of hcat2
  pack(2, 100, 100, 200, 208);
  gemm_wmma_kernel<100, 208, 13, 200><<<blk_bi, 256, 0, stream>>>(hcat1, wtp, P(2, 2), P(2, 3), xg, M);
  lstm_recur_kernel<50, 200, 52, 13, 208><<<BSEQ / 16, 256, 0, stream>>>(xg, P(2, 1), hcat2, 100, 0, 0);
  pack(3, 100, 100, 200, 208);
  gemm_wmma_kernel<100, 208, 13, 200><<<blk_bi, 256, 0, stream>>>(hcat1, wtp, P(3, 2), P(3, 3), xg, M);
  lstm_recur_kernel<50, 200, 52, 13, 208><<<BSEQ / 16, 256, 0, stream>>>(xg, P(3, 1), hcat2, 100, 50, 1);

  // 3) uni0 (in 100 -> hid 10), writes stride-12 padded h-buffer
  pack(4, 100, 100, 40, 48);
  gemm_wmma_kernel<100, 48, 3, 40><<<blk_un, 256, 0, stream>>>(hcat2, wtp, P(4, 2), P(4, 3), xg, M);
  lstm_recur_kernel<10, 40, 12, 3, 48><<<BSEQ / 16, 256, 0, stream>>>(xg, P(4, 1), hu1p, 12, 0, 0);

  // 4) uni1 (in 10 pad 12 -> hid 10)
  pack(5, 10, 12, 40, 48);
  gemm_wmma_kernel<12, 48, 3, 40><<<blk_un, 256, 0, stream>>>(hu1p, wtp, P(5, 2), P(5, 3), xg, M);
  lstm_recur_kernel<10, 40, 12, 3, 48><<<BSEQ / 16, 256, 0, stream>>>(xg, P(5, 1), hu2, 10, 0, 0);

  // 5) last timestep -> NCHW output
  extract_out_kernel<<<(8 * 10 * 400 + 255) / 256, 256, 0, stream>>>(hu2, (float*)d_out);
}